// EdgeWeightAndSum_v2_4174708212119
// MI455X (gfx1250) — compile-verified
//
#include <hip/hip_runtime.h>
#include <hip/hip_bf16.h>
#include <math.h>

#define E_TOTAL   1600000
#define N_NODES   100000
#define N_GRAPHS  128
#define FDIM      64
#define NEG_SLOPE 0.01f

typedef float v2f __attribute__((ext_vector_type(2)));
typedef float v8f __attribute__((ext_vector_type(8)));

// ---- monotonic order-preserving float<->uint key (for atomic max on f32) ----
__device__ __forceinline__ unsigned f2key(float f) {
    unsigned i = __float_as_uint(f);
    return (i & 0x80000000u) ? ~i : (i | 0x80000000u);
}
__device__ __forceinline__ float key2f(unsigned k) {
    unsigned i = (k & 0x80000000u) ? (k & 0x7FFFFFFFu) : ~k;
    return __uint_as_float(i);
}

// ---- init: m-keys to 0 (== -inf under key order), den to 0, h_g output to 0 ----
__global__ void init_kernel(unsigned* __restrict__ mkeys,
                            float* __restrict__ den,
                            float* __restrict__ out_hg) {
    int i = blockIdx.x * blockDim.x + threadIdx.x;
    if (i < N_NODES) { mkeys[i] = 0u; den[i] = 0.0f; }
    if (i < N_GRAPHS * FDIM) out_hg[i] = 0.0f;
}

// ---- per-edge score via f32 WMMA: s = leaky_relu(feats @ W + b) ----
// One wave handles a 16-edge tile; 16 chained V_WMMA_F32_16X16X4_F32 ops.
// A (16x4, f32): lane<16 -> row M=lane, v0/v1 = K0/K1 ; lane>=16 -> K2/K3.
// B (4x16): W broadcast across all 16 columns -> D[m][n] = dot(feats[m], W) for all n.
__global__ void __launch_bounds__(256)
score_kernel(const float* __restrict__ feats,
             const float* __restrict__ W,
             const float* __restrict__ bias_p,
             float* __restrict__ s) {
    const int lane = threadIdx.x & 31;
    const int wave = threadIdx.x >> 5;
    const int e0   = (blockIdx.x * 8 + wave) * 16;
    const int row  = lane & 15;   // edge within tile (A-matrix M)
    const int kh   = lane >> 4;   // K half: 0 -> K0,K1 ; 1 -> K2,K3

    const float* rowp = feats + (size_t)(e0 + row) * FDIM + 2 * kh;
    const float* wp   = W + 2 * kh;

    v8f acc = {0.f, 0.f, 0.f, 0.f, 0.f, 0.f, 0.f, 0.f};
#pragma unroll
    for (int kb = 0; kb < FDIM; kb += 4) {
        const float2 av = *(const float2*)(rowp + kb);
        const float2 wv = *(const float2*)(wp + kb);
        v2f a; a.x = av.x; a.y = av.y;
        v2f b; b.x = wv.x; b.y = wv.y;
        acc = __builtin_amdgcn_wmma_f32_16x16x4_f32(
            false, a, false, b, (short)0, acc, false, false);
    }

    // D layout: VGPR r, lanes 0-15 -> M=r ; lanes 16-31 -> M=8+r (N irrelevant: broadcast)
    if (row == 0) {
        const float bv = bias_p[0];
        float* op = s + e0 + 8 * kh;
        float4 lo, hi;
        {
            float v0 = acc[0] + bv, v1 = acc[1] + bv, v2 = acc[2] + bv, v3 = acc[3] + bv;
            lo.x = (v0 >= 0.f) ? v0 : NEG_SLOPE * v0;
            lo.y = (v1 >= 0.f) ? v1 : NEG_SLOPE * v1;
            lo.z = (v2 >= 0.f) ? v2 : NEG_SLOPE * v2;
            lo.w = (v3 >= 0.f) ? v3 : NEG_SLOPE * v3;
        }
        {
            float v0 = acc[4] + bv, v1 = acc[5] + bv, v2 = acc[6] + bv, v3 = acc[7] + bv;
            hi.x = (v0 >= 0.f) ? v0 : NEG_SLOPE * v0;
            hi.y = (v1 >= 0.f) ? v1 : NEG_SLOPE * v1;
            hi.z = (v2 >= 0.f) ? v2 : NEG_SLOPE * v2;
            hi.w = (v3 >= 0.f) ? v3 : NEG_SLOPE * v3;
        }
        *(float4*)(op)     = lo;
        *(float4*)(op + 4) = hi;
    }
}

// ---- segmented max over sorted edge_dst (wave segmented scan + atomicMax on keys) ----
__global__ void __launch_bounds__(256)
segmax_kernel(const float* __restrict__ s,
              const int* __restrict__ dst,
              unsigned* __restrict__ mkeys) {
    const int e = blockIdx.x * blockDim.x + threadIdx.x;
    const int lane = threadIdx.x & 31;
    const int d = dst[e];
    unsigned key = f2key(s[e]);
#pragma unroll
    for (int off = 1; off < 32; off <<= 1) {
        unsigned ok = __shfl_up(key, off, 32);
        int      od = __shfl_up(d,   off, 32);
        if (lane >= off && od == d && ok > key) key = ok;
    }
    const int nd = __shfl_down(d, 1, 32);
    if (lane == 31 || nd != d) atomicMax(mkeys + d, key);
}

// ---- ex = exp(s - m[dst]); write ex into weights slot; segmented sum -> den ----
__global__ void __launch_bounds__(256)
expsum_kernel(const float* __restrict__ s,
              const int* __restrict__ dst,
              const unsigned* __restrict__ mkeys,
              float* __restrict__ exw,
              float* __restrict__ den) {
    const int e = blockIdx.x * blockDim.x + threadIdx.x;
    const int lane = threadIdx.x & 31;
    const int d = dst[e];
    const float m = key2f(mkeys[d]);
    const float ex = expf(s[e] - m);
    exw[e] = ex;
    float sum = ex;
#pragma unroll
    for (int off = 1; off < 32; off <<= 1) {
        float os = __shfl_up(sum, off, 32);
        int   od = __shfl_up(d,   off, 32);
        if (lane >= off && od == d) sum += os;
    }
    const int nd = __shfl_down(d, 1, 32);
    if (lane == 31 || nd != d) atomicAdd(den + d, sum);
}

// ---- normalize weights in place: w = ex / den[dst] ----
__global__ void __launch_bounds__(256)
normw_kernel(float* __restrict__ wts,
             const int* __restrict__ dst,
             const float* __restrict__ den) {
    const int e = blockIdx.x * blockDim.x + threadIdx.x;
    wts[e] = wts[e] / den[dst[e]];
}

// ---- per-graph weighted feature sum (sorted graph_ids) ----
// 64 threads = one feature column each; 512-edge chunk staged through LDS;
// register accumulation, atomicAdd only on graph transitions.
#define CHUNK 512
__global__ void __launch_bounds__(64)
graphsum_kernel(const float* __restrict__ feats,
                const float* __restrict__ wts,
                const int* __restrict__ gid,
                float* __restrict__ out) {
    __shared__ float sw[CHUNK];
    __shared__ int   sg[CHUNK];
    const int f = threadIdx.x;             // 0..63 feature column
    const int base = blockIdx.x * CHUNK;
    for (int i = f; i < CHUNK; i += 64) {
        sw[i] = wts[base + i];
        sg[i] = gid[base + i];
    }
    __syncthreads();
    float acc = 0.f;
    int gprev = sg[0];
    for (int i = 0; i < CHUNK; ++i) {
        const int g = sg[i];
        if (g != gprev) {
            atomicAdd(&out[gprev * FDIM + f], acc);
            acc = 0.f;
            gprev = g;
        }
        acc += feats[(size_t)(base + i) * FDIM + f] * sw[i];
    }
    atomicAdd(&out[gprev * FDIM + f], acc);
}

extern "C" void kernel_launch(void* const* d_in, const int* in_sizes, int n_in,
                              void* d_out, int out_size, void* d_ws, size_t ws_size,
                              hipStream_t stream) {
    const float* feats = (const float*)d_in[0];   // [E, 64]
    const float* W     = (const float*)d_in[1];   // [64, 1]
    const float* b     = (const float*)d_in[2];   // [1]
    const int*   dst   = (const int*)d_in[3];     // [E] sorted
    const int*   gid   = (const int*)d_in[4];     // [E] sorted

    float* out = (float*)d_out;                   // [0,8192): h_g_sum ; [8192, 8192+E): weights
    float* wts = out + N_GRAPHS * FDIM;

    // workspace: s[E] f32 | mkeys[N] u32 | den[N] f32   (~7.2 MB)
    float*    s     = (float*)d_ws;
    unsigned* mkeys = (unsigned*)((char*)d_ws + (size_t)E_TOTAL * 4);
    float*    den   = (float*)((char*)d_ws + (size_t)E_TOTAL * 4 + (size_t)N_NODES * 4);

    init_kernel   <<<(N_NODES + 255) / 256, 256, 0, stream>>>(mkeys, den, out);
    score_kernel  <<<E_TOTAL / 128,        256, 0, stream>>>(feats, W, b, s);
    segmax_kernel <<<E_TOTAL / 256,        256, 0, stream>>>(s, dst, mkeys);
    expsum_kernel <<<E_TOTAL / 256,        256, 0, stream>>>(s, dst, mkeys, wts, den);
    normw_kernel  <<<E_TOTAL / 256,        256, 0, stream>>>(wts, dst, den);
    graphsum_kernel<<<E_TOTAL / CHUNK,      64, 0, stream>>>(feats, wts, gid, out);
}